// SensorGroupAttention_67319317398088
// MI455X (gfx1250) — compile-verified
//
#include <hip/hip_runtime.h>
#include <hip/hip_bf16.h>

#define Bn 64
#define Cn 20
#define Tn 32768
#define Gn 5
#define Sn 4
#define En 32
#define Hn 4
#define Kn 3
#define HDn 8
#define BN_EPS 1e-5f

typedef __attribute__((ext_vector_type(2))) float v2f;
typedef __attribute__((ext_vector_type(8))) float v8f;

__global__ void init_ws_kernel(float* ws) {
    int i = blockIdx.x * blockDim.x + threadIdx.x;
    if (i < Bn * Gn * Sn) ws[i] = 0.f;
}

// ---------------- Pass 1: conv + BN + ReLU + pooled sums ----------------
__global__ __launch_bounds__(256) void conv_pool_kernel(
    const float* __restrict__ x, const int* __restrict__ gidx,
    const float* __restrict__ conv_w,
    const float* __restrict__ bng, const float* __restrict__ bnb,
    const float* __restrict__ bnm, const float* __restrict__ bnv,
    float* __restrict__ pooled)
{
    const int bg = blockIdx.y;
    const int b = bg / Gn, g = bg % Gn;
    const int tid = threadIdx.x;

    __shared__ float wgt[Sn][Sn][Kn];
    __shared__ float scale[Sn], shift[Sn];
    __shared__ int chs[Sn];
    __shared__ float red[Sn][8];

    if (tid < Sn * Sn * Kn) ((float*)wgt)[tid] = conv_w[g * Sn * Sn * Kn + tid];
    if (tid < Sn) {
        float sc = bng[g * Sn + tid] * rsqrtf(bnv[g * Sn + tid] + BN_EPS);
        scale[tid] = sc;
        shift[tid] = bnb[g * Sn + tid] - bnm[g * Sn + tid] * sc;
        chs[tid] = gidx[g * Sn + tid];
    }
    __syncthreads();

    const int t0 = (blockIdx.x * 256 + tid) * 4;

    float win[Sn][6];
    #pragma unroll
    for (int si = 0; si < Sn; ++si) {
        const float* xc = x + ((size_t)b * Cn + chs[si]) * Tn;
        float4 v = *(const float4*)(xc + t0);
        win[si][1] = v.x; win[si][2] = v.y; win[si][3] = v.z; win[si][4] = v.w;
        win[si][0] = (t0 > 0) ? xc[t0 - 1] : 0.f;
        win[si][5] = (t0 + 4 < Tn) ? xc[t0 + 4] : 0.f;
    }

    float ls[Sn];
    #pragma unroll
    for (int s = 0; s < Sn; ++s) {
        float a0 = 0.f, a1 = 0.f, a2 = 0.f, a3 = 0.f;
        #pragma unroll
        for (int si = 0; si < Sn; ++si) {
            #pragma unroll
            for (int k = 0; k < Kn; ++k) {
                float w = wgt[s][si][k];
                a0 = fmaf(w, win[si][0 + k], a0);
                a1 = fmaf(w, win[si][1 + k], a1);
                a2 = fmaf(w, win[si][2 + k], a2);
                a3 = fmaf(w, win[si][3 + k], a3);
            }
        }
        float f0 = fmaxf(0.f, fmaf(a0, scale[s], shift[s]));
        float f1 = fmaxf(0.f, fmaf(a1, scale[s], shift[s]));
        float f2 = fmaxf(0.f, fmaf(a2, scale[s], shift[s]));
        float f3 = fmaxf(0.f, fmaf(a3, scale[s], shift[s]));
        ls[s] = (f0 + f1) + (f2 + f3);
    }

    // wave32 reduction, then cross-wave via LDS, then 4 atomics per block
    #pragma unroll
    for (int s = 0; s < Sn; ++s) {
        float v = ls[s];
        #pragma unroll
        for (int off = 16; off > 0; off >>= 1) v += __shfl_down(v, off, 32);
        if ((tid & 31) == 0) red[s][tid >> 5] = v;
    }
    __syncthreads();
    if (tid < Sn) {
        float v = 0.f;
        #pragma unroll
        for (int w = 0; w < 8; ++w) v += red[tid][w];
        atomicAdd(&pooled[(b * Gn + g) * Sn + tid], v);
    }
}

// ---------------- Pass 2: tokens -> attention -> gate (WMMA f32) ----------------
__global__ __launch_bounds__(32) void attn_gate_kernel(
    const float* __restrict__ pooled,
    const float* __restrict__ tok_w, const float* __restrict__ tok_b,
    const float* __restrict__ gate_w, const float* __restrict__ gate_b,
    const float* __restrict__ ipw, const float* __restrict__ ipb,
    const float* __restrict__ opw, const float* __restrict__ opb,
    float* __restrict__ gate_out)
{
    const int b = blockIdx.x;
    const int lane = threadIdx.x;

    __shared__ float tok[16][En];   // rows >= Gn are zero
    __shared__ float qm[Gn][En], km[Gn][En], vm[Gn][En];
    __shared__ float ctxm[16][En];  // rows >= Gn are zero
    __shared__ float attm[Gn][En];

    const float invT = 1.0f / (float)Tn;
    #pragma unroll
    for (int m = 0; m < 16; ++m) {
        float v = 0.f;
        if (m < Gn) {
            float p0 = pooled[(b * Gn + m) * Sn + 0] * invT;
            float p1 = pooled[(b * Gn + m) * Sn + 1] * invT;
            float p2 = pooled[(b * Gn + m) * Sn + 2] * invT;
            float p3 = pooled[(b * Gn + m) * Sn + 3] * invT;
            const float* tw = tok_w + ((size_t)m * En + lane) * Sn;
            v = tok_b[m * En + lane] + p0 * tw[0] + p1 * tw[1] + p2 * tw[2] + p3 * tw[3];
        }
        tok[m][lane] = v;
        ctxm[m][lane] = 0.f;
    }
    __syncthreads();

    const int mrow  = lane & 15;
    const int khalf = (lane >> 4) << 1;   // 0 for lanes 0-15, 2 for lanes 16-31
    const int ncol  = lane & 15;

    // q/k/v projections: tokens(16x32) @ W^T tile -> 16x16, K in 8 steps of 4
    #pragma unroll
    for (int p = 0; p < 3; ++p) {
        #pragma unroll
        for (int nt = 0; nt < 2; ++nt) {
            v8f acc = {};
            #pragma unroll
            for (int k0 = 0; k0 < En; k0 += 4) {
                v2f a, bb;
                a.x = tok[mrow][k0 + khalf];
                a.y = tok[mrow][k0 + khalf + 1];
                int j = p * En + nt * 16 + ncol;
                bb.x = ipw[(size_t)j * En + k0 + khalf];
                bb.y = ipw[(size_t)j * En + k0 + khalf + 1];
                acc = __builtin_amdgcn_wmma_f32_16x16x4_f32(
                    false, a, false, bb, (short)0, acc, false, false);
            }
            if (lane < 16) {  // lanes 0-15 hold rows M=0..7 in acc[0..7]
                float bias = ipb[p * En + nt * 16 + ncol];
                float* dst = (p == 0) ? &qm[0][0] : (p == 1) ? &km[0][0] : &vm[0][0];
                #pragma unroll
                for (int i = 0; i < Gn; ++i)
                    dst[i * En + nt * 16 + ncol] = acc[i] + bias;
            }
        }
    }
    __syncthreads();

    // attention: lanes 0..19 each own (head h, query row qi)
    if (lane < Hn * Gn) {
        int h = lane / Gn, qi = lane % Gn;
        float sr[Gn];
        float mx = -1e30f;
        #pragma unroll
        for (int ki = 0; ki < Gn; ++ki) {
            float s = 0.f;
            #pragma unroll
            for (int d = 0; d < HDn; ++d) s += qm[qi][h * HDn + d] * km[ki][h * HDn + d];
            s *= 0.35355339059327373f;  // 1/sqrt(8)
            sr[ki] = s; mx = fmaxf(mx, s);
        }
        float sum = 0.f;
        #pragma unroll
        for (int ki = 0; ki < Gn; ++ki) { sr[ki] = __expf(sr[ki] - mx); sum += sr[ki]; }
        float inv = 1.0f / sum;
        #pragma unroll
        for (int d = 0; d < HDn; ++d) {
            float c = 0.f;
            #pragma unroll
            for (int ki = 0; ki < Gn; ++ki) c += sr[ki] * vm[ki][h * HDn + d];
            ctxm[qi][h * HDn + d] = c * inv;
        }
    }
    __syncthreads();

    // out projection: ctx(16x32) @ out_w^T
    #pragma unroll
    for (int nt = 0; nt < 2; ++nt) {
        v8f acc = {};
        #pragma unroll
        for (int k0 = 0; k0 < En; k0 += 4) {
            v2f a, bb;
            a.x = ctxm[mrow][k0 + khalf];
            a.y = ctxm[mrow][k0 + khalf + 1];
            int j = nt * 16 + ncol;
            bb.x = opw[(size_t)j * En + k0 + khalf];
            bb.y = opw[(size_t)j * En + k0 + khalf + 1];
            acc = __builtin_amdgcn_wmma_f32_16x16x4_f32(
                false, a, false, bb, (short)0, acc, false, false);
        }
        if (lane < 16) {
            float bias = opb[nt * 16 + ncol];
            #pragma unroll
            for (int i = 0; i < Gn; ++i)
                attm[i][nt * 16 + ncol] = acc[i] + bias;
        }
    }
    __syncthreads();

    // gate = sigmoid(attended @ gate_w^T + gate_b), one lane per (g,s)
    if (lane < Gn * Sn) {
        int g = lane / Sn, s = lane % Sn;
        float z = gate_b[g * Sn + s];
        const float* gw = gate_w + ((size_t)g * Sn + s) * En;
        #pragma unroll
        for (int e = 0; e < En; ++e) z += attm[g][e] * gw[e];
        gate_out[b * Gn * Sn + lane] = 1.0f / (1.0f + __expf(-z));
    }
}

// ---------------- Pass 3: recompute conv+BN+ReLU, gate, residual ----------------
__global__ __launch_bounds__(256) void conv_apply_kernel(
    const float* __restrict__ x, const int* __restrict__ gidx,
    const float* __restrict__ conv_w,
    const float* __restrict__ bng, const float* __restrict__ bnb,
    const float* __restrict__ bnm, const float* __restrict__ bnv,
    const float* __restrict__ gate, const float* __restrict__ alpha_p,
    float* __restrict__ out)
{
    const int bg = blockIdx.y;
    const int b = bg / Gn, g = bg % Gn;
    const int tid = threadIdx.x;

    __shared__ float wgt[Sn][Sn][Kn];
    __shared__ float scale[Sn], shift[Sn], ag[Sn];  // ag = alpha * gate
    __shared__ int chs[Sn];

    if (tid < Sn * Sn * Kn) ((float*)wgt)[tid] = conv_w[g * Sn * Sn * Kn + tid];
    if (tid < Sn) {
        float sc = bng[g * Sn + tid] * rsqrtf(bnv[g * Sn + tid] + BN_EPS);
        scale[tid] = sc;
        shift[tid] = bnb[g * Sn + tid] - bnm[g * Sn + tid] * sc;
        chs[tid] = gidx[g * Sn + tid];
        ag[tid] = alpha_p[0] * gate[(b * Gn + g) * Sn + tid];
    }
    __syncthreads();

    const int t0 = (blockIdx.x * 256 + tid) * 4;

    float win[Sn][6];
    #pragma unroll
    for (int si = 0; si < Sn; ++si) {
        const float* xc = x + ((size_t)b * Cn + chs[si]) * Tn;
        float4 v = *(const float4*)(xc + t0);
        win[si][1] = v.x; win[si][2] = v.y; win[si][3] = v.z; win[si][4] = v.w;
        win[si][0] = (t0 > 0) ? xc[t0 - 1] : 0.f;
        win[si][5] = (t0 + 4 < Tn) ? xc[t0 + 4] : 0.f;
    }

    #pragma unroll
    for (int s = 0; s < Sn; ++s) {
        float a0 = 0.f, a1 = 0.f, a2 = 0.f, a3 = 0.f;
        #pragma unroll
        for (int si = 0; si < Sn; ++si) {
            #pragma unroll
            for (int k = 0; k < Kn; ++k) {
                float w = wgt[s][si][k];
                a0 = fmaf(w, win[si][0 + k], a0);
                a1 = fmaf(w, win[si][1 + k], a1);
                a2 = fmaf(w, win[si][2 + k], a2);
                a3 = fmaf(w, win[si][3 + k], a3);
            }
        }
        float f0 = fmaxf(0.f, fmaf(a0, scale[s], shift[s]));
        float f1 = fmaxf(0.f, fmaf(a1, scale[s], shift[s]));
        float f2 = fmaxf(0.f, fmaf(a2, scale[s], shift[s]));
        float f3 = fmaxf(0.f, fmaf(a3, scale[s], shift[s]));
        float m = ag[s];
        float4 o;
        // residual x value is already in the window (channel chs[s] == window si=s)
        o.x = fmaf(m, f0, win[s][1]);
        o.y = fmaf(m, f1, win[s][2]);
        o.z = fmaf(m, f2, win[s][3]);
        o.w = fmaf(m, f3, win[s][4]);
        *(float4*)(out + ((size_t)b * Cn + chs[s]) * Tn + t0) = o;
    }
}

extern "C" void kernel_launch(void* const* d_in, const int* in_sizes, int n_in,
                              void* d_out, int out_size, void* d_ws, size_t ws_size,
                              hipStream_t stream) {
    const float* x      = (const float*)d_in[0];
    const int*   gidx   = (const int*)  d_in[1];
    const float* conv_w = (const float*)d_in[2];
    const float* bng    = (const float*)d_in[3];
    const float* bnb    = (const float*)d_in[4];
    const float* bnm    = (const float*)d_in[5];
    const float* bnv    = (const float*)d_in[6];
    const float* tok_w  = (const float*)d_in[7];
    const float* tok_b  = (const float*)d_in[8];
    const float* gate_w = (const float*)d_in[9];
    const float* gate_b = (const float*)d_in[10];
    const float* ipw    = (const float*)d_in[11];
    const float* ipb    = (const float*)d_in[12];
    const float* opw    = (const float*)d_in[13];
    const float* opb    = (const float*)d_in[14];
    const float* alpha  = (const float*)d_in[15];
    float* out = (float*)d_out;

    float* pooled = (float*)d_ws;            // B*G*S = 1280 floats
    float* gate   = pooled + Bn * Gn * Sn;   // B*G*S = 1280 floats

    init_ws_kernel<<<(Bn * Gn * Sn + 255) / 256, 256, 0, stream>>>(pooled);

    dim3 grid1(Tn / (256 * 4), Bn * Gn);
    conv_pool_kernel<<<grid1, 256, 0, stream>>>(x, gidx, conv_w, bng, bnb, bnm, bnv, pooled);

    attn_gate_kernel<<<Bn, 32, 0, stream>>>(pooled, tok_w, tok_b, gate_w, gate_b,
                                            ipw, ipb, opw, opb, gate);

    conv_apply_kernel<<<grid1, 256, 0, stream>>>(x, gidx, conv_w, bng, bnb, bnm, bnv,
                                                 gate, alpha, out);
}